// GraphSAGEConv_83408264888608
// MI455X (gfx1250) — compile-verified
//
#include <hip/hip_runtime.h>

typedef __attribute__((ext_vector_type(2))) float v2f;
typedef __attribute__((ext_vector_type(8))) float v8f;

#define IN_DIM  128
#define OUT_DIM 128
#define KDIM    256   // 2*IN_DIM
#define KCHUNK  64    // W rows staged per LDS chunk (64*128*4 = 32KB)

// ---------------- zero init: deg (ws) and agg (aliases d_out) ----------------
__global__ void zero_kernel(float* __restrict__ deg, float* __restrict__ agg,
                            int ndeg, long nagg) {
    long i = (long)blockIdx.x * blockDim.x + threadIdx.x;
    if (i < ndeg) deg[i] = 0.0f;
    if (i < nagg) agg[i] = 0.0f;
}

// ---------------- out-degree of src ----------------
__global__ void degree_kernel(const int* __restrict__ src, float* __restrict__ deg,
                              int nedges) {
    int e = blockIdx.x * blockDim.x + threadIdx.x;
    if (e < nedges) atomicAdd(&deg[src[e]], 1.0f);
}

// ---------------- deg -> 1/max(deg,1) in place ----------------
__global__ void invdeg_kernel(float* __restrict__ deg, int n) {
    int i = blockIdx.x * blockDim.x + threadIdx.x;
    if (i < n) deg[i] = 1.0f / fmaxf(deg[i], 1.0f);
}

// ---------------- edge scatter: agg[dst] += x[src] * invdeg[src] -------------
// one wave per edge: 32 lanes x float4 = 128 features, coalesced 512B row read
__global__ void scatter_kernel(const float* __restrict__ x,
                               const int* __restrict__ src,
                               const int* __restrict__ dst,
                               const float* __restrict__ invdeg,
                               float* __restrict__ agg, int nedges) {
    int t = blockIdx.x * blockDim.x + threadIdx.x;
    int e = t >> 5;
    if (e >= nedges) return;
    int lane = t & 31;
    int s = src[e];
    int d = dst[e];
    float w = invdeg[s];
    const float4* xr = (const float4*)(x + (size_t)s * IN_DIM);
    float4 v = xr[lane];
    float* ar = agg + (size_t)d * IN_DIM + lane * 4;
    atomicAdd(ar + 0, v.x * w);
    atomicAdd(ar + 1, v.y * w);
    atomicAdd(ar + 2, v.z * w);
    atomicAdd(ar + 3, v.w * w);
}

// LDS address for W element (k within chunk, n): k-pair interleaved + XOR-16
// bank swizzle. Pair (k, k+1) at column n is one contiguous 8B element.
__device__ __forceinline__ int wlds_idx(int k, int n) {
    int p = k >> 1;
    int ns = n ^ ((p & 1) << 4);
    return p * (2 * OUT_DIM) + ns * 2 + (k & 1);
}

// ---------------- GEMM: out = relu([x | agg] @ W + b), agg aliases out -------
// wave computes 16x128 strip via 8 f32 accumulators; W staged in LDS chunks
__global__ __launch_bounds__(256)
void gemm_kernel(const float* __restrict__ x,
                 const float* __restrict__ W,      // [256,128] row-major
                 const float* __restrict__ bias,   // [128]
                 float* __restrict__ out,          // [N,128]; K>=128 reads = agg
                 int N) {
    __shared__ float Wlds[KCHUNK * OUT_DIM];   // 32 KB, pair-interleaved

    const int tid  = threadIdx.x;
    const int wave = tid >> 5;                 // 0..7
    const int lane = tid & 31;

    const int mtile = blockIdx.x * 8 + wave;
    const int m0    = mtile * 16;
    const bool active = (m0 < N);              // N % 16 == 0 in this problem

    const int mrow = active ? (m0 + (lane & 15)) : 0;
    const int koff = (lane < 16) ? 0 : 2;      // -> pair parity 0 / 1
    const int ncol = lane & 15;

    // per-lane A base pointers, already offset by this lane's koff
    const float* xrow = x   + (size_t)mrow * IN_DIM  + koff;
    const float* grow = out + (size_t)mrow * OUT_DIM + koff;  // agg row

    v8f acc[8] = {};

    for (int kc = 0; kc < KDIM; kc += KCHUNK) {
        __syncthreads();
        // stage W rows [kc, kc+KCHUNK) into LDS (pair-interleaved, swizzled):
        // each thread copies float2 of consecutive n, 2 scalar LDS stores
        {
            const float2* Wv = (const float2*)(W + (size_t)kc * OUT_DIM);
            for (int i = tid; i < (KCHUNK * OUT_DIM) / 2; i += 256) {
                int k = i >> 6;            // 0..KCHUNK-1
                int n = (i & 63) * 2;      // even column
                float2 v = Wv[i];
                Wlds[wlds_idx(k, n)]     = v.x;
                Wlds[wlds_idx(k, n + 1)] = v.y;
            }
        }
        __syncthreads();

        // chunk-invariant A row pointer (select hoisted out of the k loop so
        // loads become base + immediate offset)
        const float* srow = (kc < IN_DIM) ? (xrow + kc) : (grow + (kc - IN_DIM));

        for (int kk = 0; kk < KCHUNK; kk += 4) {
            v2f a = *(const v2f*)(srow + kk);   // global_load_b64, imm offset
            // pair index for this lane: p = (kk+koff)/2, parity = koff/2
            const int p  = (kk + koff) >> 1;
            const float* prow = &Wlds[p * (2 * OUT_DIM)];
            const int nsw = ((p & 1) << 4);
#pragma unroll
            for (int nt = 0; nt < 8; ++nt) {
                const int n = (nt * 16 + ncol) ^ nsw;
                v2f bb = *(const v2f*)(prow + n * 2);   // single ds_load_b64
                acc[nt] = __builtin_amdgcn_wmma_f32_16x16x4_f32(
                    /*neg_a=*/false, a, /*neg_b=*/false, bb,
                    /*c_mod=*/(short)0, acc[nt],
                    /*reuse_a=*/false, /*reuse_b=*/false);
            }
        }
    }

    if (!active) return;

    // epilogue: D VGPR i -> row m0 + i + (lane<16 ? 0 : 8), col = nt*16 + ncol
    const int rbase = m0 + ((lane < 16) ? 0 : 8);
#pragma unroll
    for (int nt = 0; nt < 8; ++nt) {
        const int n  = nt * 16 + ncol;
        const float bv = bias[n];
#pragma unroll
        for (int i = 0; i < 8; ++i) {
            float v = acc[nt][i] + bv;
            out[(size_t)(rbase + i) * OUT_DIM + n] = fmaxf(v, 0.0f);
        }
    }
}

extern "C" void kernel_launch(void* const* d_in, const int* in_sizes, int n_in,
                              void* d_out, int out_size, void* d_ws, size_t ws_size,
                              hipStream_t stream) {
    const float* x    = (const float*)d_in[0];
    const int*   src  = (const int*)  d_in[1];
    const int*   dst  = (const int*)  d_in[2];
    const float* W    = (const float*)d_in[3];
    const float* bias = (const float*)d_in[4];
    float* out = (float*)d_out;

    const int N = in_sizes[0] / IN_DIM;   // 100000
    const int E = in_sizes[1];            // 1600000

    float* deg = (float*)d_ws;            // N floats of scratch
    float* agg = out;                     // agg aliases the output buffer
    const long nagg = (long)N * IN_DIM;

    // 1) zero deg + agg
    {
        int blocks = (int)((nagg + 255) / 256);
        zero_kernel<<<blocks, 256, 0, stream>>>(deg, agg, N, nagg);
    }
    // 2) out-degree of src
    degree_kernel<<<(E + 255) / 256, 256, 0, stream>>>(src, deg, E);
    // 3) invert degree
    invdeg_kernel<<<(N + 255) / 256, 256, 0, stream>>>(deg, N);
    // 4) scatter messages (one wave per edge)
    {
        long threads = (long)E * 32;
        int blocks = (int)((threads + 255) / 256);
        scatter_kernel<<<blocks, 256, 0, stream>>>(x, src, dst, deg, agg, E);
    }
    // 5) fused concat-GEMM + bias + relu via f32 WMMA (in-place over agg)
    {
        int mtiles = (N + 15) / 16;             // 6250
        int blocks = (mtiles + 7) / 8;          // 8 waves/block, 1 Mtile/wave
        gemm_kernel<<<blocks, 256, 0, stream>>>(x, W, bias, out, N);
    }
}